// InsulationScoreModule_88682484727889
// MI455X (gfx1250) — compile-verified
//
#include <hip/hip_runtime.h>

// CDNA5 / gfx1250, wave32. f32 WMMA used as a ones-matmul reduction engine.

typedef __attribute__((ext_vector_type(2))) float v2f;
typedef __attribute__((ext_vector_type(8))) float v8f;

#define NDIM 4096
#define WIN  10
#define NOUT (NDIM - 2 * WIN)        // 4076 outputs per batch
#define NTILES ((NOUT + 15) / 16)    // 255 16-output tiles per batch
#define WAVES_PER_BLOCK 4

__global__ __launch_bounds__(128)
void insulation_score_wmma(const float* __restrict__ x, float* __restrict__ out) {
    const int lane  = threadIdx.x & 31;
    const int wave  = threadIdx.x >> 5;
    const int tile  = blockIdx.x * WAVES_PER_BLOCK + wave;  // 0..255 (256 launched, 255 used)
    const int b     = blockIdx.y;                            // batch 0..3
    const int m     = lane & 15;                             // A-matrix row M = lane % 16
    const int khalf = lane >> 4;                             // 0: K=0,1  1: K=2,3 (per 16x4 f32 A layout)

    // Output index this lane's A-row corresponds to; clamp tail lanes so all
    // loads stay in-bounds (their D rows are discarded at the store).
    int i  = tile * 16 + m;
    int ii = (i < NOUT) ? i : (NOUT - 1);

    // Window top-left for output ii: row = ii, col = ii + 11.
    const float* p = x + (size_t)b * NDIM * NDIM + (size_t)ii * NDIM + (size_t)(ii + WIN + 1);

    v8f acc = {};                 // 16x16 f32 accumulator (8 VGPRs, wave32)
    v2f ones; ones.x = 1.0f; ones.y = 1.0f;  // B = all-ones 4x16 -> D[m][*] = row sums

    // 25 chained V_WMMA_F32_16X16X4_F32: K-chunks of 4 cover kk = 0..99
    // (window element kk -> r = kk/10, c = kk%10). kk0 = 4s + 2*khalf is even,
    // so c <= 8 and the pair (kk0, kk0+1) is always column-adjacent in memory.
    #pragma unroll
    for (int s = 0; s < 25; ++s) {
        int kk = (s << 2) + (khalf << 1);
        int r  = kk / 10;
        int c  = kk - r * 10;
        const float* q = p + r * NDIM + c;
        v2f a;
        a.x = q[0];
        a.y = q[1];
        // 8 args: (neg_a, A, neg_b, B, c_mod, C, reuse_a, reuse_b)
        acc = __builtin_amdgcn_wmma_f32_16x16x4_f32(
            false, a, false, ones, (short)0, acc, false, false);
    }

    // D layout (32-bit 16x16 C/D): VGPR j, lanes 0-15 -> M=j, lanes 16-31 -> M=j+8.
    // Every N column is identical (B was ones); read column N=0 via lanes 0 and 16.
    if ((lane & 15) == 0) {
        const int mbase = khalf * 8;
        #pragma unroll
        for (int j = 0; j < 8; ++j) {
            int oi = tile * 16 + mbase + j;
            if (oi < NOUT)
                out[(size_t)b * NOUT + oi] = acc[j] * (1.0f / (WIN * WIN));
        }
    }
}

extern "C" void kernel_launch(void* const* d_in, const int* in_sizes, int n_in,
                              void* d_out, int out_size, void* d_ws, size_t ws_size,
                              hipStream_t stream) {
    const float* x = (const float*)d_in[0];   // [4, 1, 4096, 4096] f32
    float* out = (float*)d_out;               // [4, 1, 4076] f32
    dim3 grid((NTILES + WAVES_PER_BLOCK - 1) / WAVES_PER_BLOCK, 4);  // (64, 4)
    insulation_score_wmma<<<grid, 32 * WAVES_PER_BLOCK, 0, stream>>>(x, out);
}